// GraphModel_34402688041621
// MI455X (gfx1250) — compile-verified
//
#include <hip/hip_runtime.h>

#define D_FEAT 300
#define H_HEADS 2
#define C_TOT 600           // H * D
#define NEG_SLOPE 0.2f

typedef __attribute__((ext_vector_type(2))) float v2f;
typedef __attribute__((ext_vector_type(8))) float v8f;

// ---------------------------------------------------------------------------
// Kernel 1: xp[n, h*300+e] = sum_d x[n,d] * W[d,h,e]
// [20000x300] x [300x600] GEMM via V_WMMA_F32_16X16X4_F32 (f32, K=4/step).
// One wave -> one 16x16 output tile. 8 waves/block cover 128 columns.
// ---------------------------------------------------------------------------
__global__ __launch_bounds__(256) void proj_wmma(const float* __restrict__ x,
                                                 const float* __restrict__ Wf,
                                                 float* __restrict__ xp,
                                                 int N) {
  const int lane = threadIdx.x & 31;
  const int wave = threadIdx.x >> 5;
  const int row0 = blockIdx.x * 16;                 // N divisible by 16
  const int col0 = blockIdx.y * 128 + wave * 16;
  if (col0 >= C_TOT) return;                        // wave-uniform skip

  const int hi   = (lane >= 16) ? 1 : 0;
  const int m    = row0 + (lane & 15);              // A row for this lane
  const int coll = col0 + (lane & 15);              // B/C column for this lane
  const int colc = (coll < C_TOT) ? coll : (C_TOT - 1);  // clamp for loads

  v8f acc = {0.f, 0.f, 0.f, 0.f, 0.f, 0.f, 0.f, 0.f};
  const float* xrow = x + (size_t)m * D_FEAT;

  for (int k = 0; k < D_FEAT; k += 4) {
    const int kk = k + 2 * hi;                      // lanes 16-31 hold K=+2,+3
    v2f a, b;
    a.x = xrow[kk];
    a.y = xrow[kk + 1];
    b.x = Wf[(size_t)kk * C_TOT + colc];
    b.y = Wf[(size_t)(kk + 1) * C_TOT + colc];
    acc = __builtin_amdgcn_wmma_f32_16x16x4_f32(
        /*neg_a=*/false, a, /*neg_b=*/false, b,
        /*c_mod=*/(short)0, acc, /*reuse_a=*/false, /*reuse_b=*/false);
  }

  if (coll < C_TOT) {                               // guarded store (post-WMMA)
    float* op = xp + (size_t)(row0 + 8 * hi) * C_TOT + coll;
#pragma unroll
    for (int i = 0; i < 8; ++i)
      op[(size_t)i * C_TOT] = acc[i];
  }
}

// ---------------------------------------------------------------------------
// Kernel 2: a_src[n,h] = dot(xp[n,h,:], att_src[h,:]);  same for a_dst.
// One wave32 per (n,h); float4 loads (all bases 16-B aligned), shfl_xor tree.
// ---------------------------------------------------------------------------
__global__ __launch_bounds__(256) void att_logits(const float* __restrict__ xp,
                                                  const float* __restrict__ att_s,
                                                  const float* __restrict__ att_d,
                                                  float* __restrict__ a_src,
                                                  float* __restrict__ a_dst,
                                                  int N) {
  const int wid  = blockIdx.x * (blockDim.x >> 5) + (threadIdx.x >> 5);
  const int lane = threadIdx.x & 31;
  if (wid >= N * H_HEADS) return;
  const int n = wid >> 1, h = wid & 1;

  const float4* row = (const float4*)(xp + (size_t)n * C_TOT + h * D_FEAT);
  const float4* as  = (const float4*)(att_s + h * D_FEAT);
  const float4* ad  = (const float4*)(att_d + h * D_FEAT);
  float ps = 0.f, pd = 0.f;
  for (int i = lane; i < D_FEAT / 4; i += 32) {     // 75 float4 per (n,h)
    const float4 v = row[i];
    const float4 s = as[i];
    const float4 d = ad[i];
    ps += v.x * s.x + v.y * s.y + v.z * s.z + v.w * s.w;
    pd += v.x * d.x + v.y * d.y + v.z * d.z + v.w * d.w;
  }
  for (int msk = 16; msk; msk >>= 1) {
    ps += __shfl_xor(ps, msk, 32);
    pd += __shfl_xor(pd, msk, 32);
  }
  if (lane == 0) { a_src[wid] = ps; a_dst[wid] = pd; }
}

// ---------------------------------------------------------------------------
// Zero fill, 16 B per thread-iteration (agg|emax|denom region, count % 4 == 0).
// ---------------------------------------------------------------------------
__global__ void fill_zero4(uint4* __restrict__ p, size_t n4) {
  size_t i = (size_t)blockIdx.x * blockDim.x + threadIdx.x;
  const size_t stride = (size_t)gridDim.x * blockDim.x;
  const uint4 z = make_uint4(0u, 0u, 0u, 0u);
  for (; i < n4; i += stride) p[i] = z;
}

// Order-preserving encode of f32 into u32 (so atomicMax(u32) == float max).
__device__ __forceinline__ unsigned enc_f32(float f) {
  const unsigned u = __float_as_uint(f);
  return (u & 0x80000000u) ? ~u : (u | 0x80000000u);
}
__device__ __forceinline__ float dec_f32(unsigned k) {
  const unsigned u = (k & 0x80000000u) ? (k & 0x7FFFFFFFu) : ~k;
  return __uint_as_float(u);
}

// ---------------------------------------------------------------------------
// Edge pass A: e = leaky_relu(a_src[src] + a_dst[dst]); segment max via
// encoded u32 atomicMax. Edges t >= E are the self-loops.
// ---------------------------------------------------------------------------
__global__ __launch_bounds__(256) void edge_logits(const int* __restrict__ srcv,
                                                   const int* __restrict__ dstv,
                                                   const float* __restrict__ a_src,
                                                   const float* __restrict__ a_dst,
                                                   float* __restrict__ ebuf,
                                                   unsigned* __restrict__ emax,
                                                   int E, int Et) {
  const int t = blockIdx.x * blockDim.x + threadIdx.x;
  if (t >= Et) return;
  int s, d;
  if (t < E) { s = srcv[t]; d = dstv[t]; } else { s = d = t - E; }

  const float2 vs = *(const float2*)(a_src + (size_t)s * H_HEADS);
  const float2 vd = *(const float2*)(a_dst + (size_t)d * H_HEADS);
  float e0 = vs.x + vd.x;  e0 = (e0 > 0.f) ? e0 : NEG_SLOPE * e0;
  float e1 = vs.y + vd.y;  e1 = (e1 > 0.f) ? e1 : NEG_SLOPE * e1;
  *(float2*)(ebuf + (size_t)t * H_HEADS) = make_float2(e0, e1);
  atomicMax(&emax[d * H_HEADS + 0], enc_f32(e0));
  atomicMax(&emax[d * H_HEADS + 1], enc_f32(e1));
}

// ---------------------------------------------------------------------------
// Edge pass B: ee = exp(e - emax[dst]); denom[dst] += ee (f32 atomic add).
// ---------------------------------------------------------------------------
__global__ __launch_bounds__(256) void edge_exp(const int* __restrict__ dstv,
                                                float* __restrict__ ebuf,
                                                const unsigned* __restrict__ emax,
                                                float* __restrict__ denom,
                                                int E, int Et) {
  const int t = blockIdx.x * blockDim.x + threadIdx.x;
  if (t >= Et) return;
  const int d = (t < E) ? dstv[t] : (t - E);

  const float2 e = *(const float2*)(ebuf + (size_t)t * H_HEADS);
  const float ee0 = __expf(e.x - dec_f32(emax[d * H_HEADS + 0]));
  const float ee1 = __expf(e.y - dec_f32(emax[d * H_HEADS + 1]));
  *(float2*)(ebuf + (size_t)t * H_HEADS) = make_float2(ee0, ee1);
  atomicAdd(&denom[d * H_HEADS + 0], ee0);
  atomicAdd(&denom[d * H_HEADS + 1], ee1);
}

// ---------------------------------------------------------------------------
// Edge pass C (dominant): agg[dst] += xp[src] * alpha. One block (5 waves)
// per edge; 150 lanes each load one float4 (global_load_b128) of the gathered
// row and issue 4 f32 atomic adds. xp+agg = 96 MB -> resident in 192 MB L2.
// ---------------------------------------------------------------------------
__global__ __launch_bounds__(160) void edge_scatter(const int* __restrict__ srcv,
                                                    const int* __restrict__ dstv,
                                                    const float* __restrict__ xp,
                                                    const float* __restrict__ ebuf,
                                                    const float* __restrict__ denom,
                                                    float* __restrict__ agg,
                                                    int E, int Et) {
  const int t = blockIdx.x;
  int s, d;
  if (t < E) { s = srcv[t]; d = dstv[t]; } else { s = d = t - E; }

  const float* xr = xp + (size_t)s * C_TOT;
  __builtin_prefetch(xr, 0, 0);                      // global_prefetch_b8
  const float2 ee = *(const float2*)(ebuf + (size_t)t * H_HEADS);
  const float2 dn = *(const float2*)(denom + (size_t)d * H_HEADS);
  const float a0 = ee.x / dn.x;
  const float a1 = ee.y / dn.y;

  float* ar = agg + (size_t)d * C_TOT;
  const int c = threadIdx.x * 4;                     // 150 of 160 lanes active
  if (c < C_TOT) {
    const float al = (c < D_FEAT) ? a0 : a1;         // head boundary is 4-aligned
    const float4 v = *(const float4*)(xr + c);
    atomicAdd(&ar[c + 0], v.x * al);
    atomicAdd(&ar[c + 1], v.y * al);
    atomicAdd(&ar[c + 2], v.z * al);
    atomicAdd(&ar[c + 3], v.w * al);
  }
}

// ---------------------------------------------------------------------------
// Finalize: out[n] = relu(mean_h(agg) + bias) . lin_w + lin_b. Wave per node,
// float4 loads throughout.
// ---------------------------------------------------------------------------
__global__ __launch_bounds__(256) void finalize(const float* __restrict__ agg,
                                                const float* __restrict__ bias,
                                                const float* __restrict__ lin_w,
                                                const float* __restrict__ lin_b,
                                                float* __restrict__ out,
                                                int N) {
  const int wid  = blockIdx.x * (blockDim.x >> 5) + (threadIdx.x >> 5);
  const int lane = threadIdx.x & 31;
  if (wid >= N) return;
  const float* ar = agg + (size_t)wid * C_TOT;
  float p = 0.f;
  for (int i = lane; i < D_FEAT / 4; i += 32) {      // 75 float4
    const float4 g0 = *(const float4*)(ar + 4 * i);
    const float4 g1 = *(const float4*)(ar + D_FEAT + 4 * i);
    const float4 b  = *(const float4*)(bias + 4 * i);
    const float4 w  = *(const float4*)(lin_w + 4 * i);
    float v;
    v = 0.5f * (g0.x + g1.x) + b.x; v = (v > 0.f) ? v : 0.f; p += v * w.x;
    v = 0.5f * (g0.y + g1.y) + b.y; v = (v > 0.f) ? v : 0.f; p += v * w.y;
    v = 0.5f * (g0.z + g1.z) + b.z; v = (v > 0.f) ? v : 0.f; p += v * w.z;
    v = 0.5f * (g0.w + g1.w) + b.w; v = (v > 0.f) ? v : 0.f; p += v * w.w;
  }
  for (int msk = 16; msk; msk >>= 1) p += __shfl_xor(p, msk, 32);
  if (lane == 0) out[wid] = p + lin_b[0];
}

// ---------------------------------------------------------------------------
extern "C" void kernel_launch(void* const* d_in, const int* in_sizes, int n_in,
                              void* d_out, int out_size, void* d_ws, size_t ws_size,
                              hipStream_t stream) {
  const float* x       = (const float*)d_in[0];
  const int*   ei      = (const int*)  d_in[1];   // [2, E]
  const float* Wf      = (const float*)d_in[2];   // [D, H, D] row-major = [300, 600]
  const float* att_src = (const float*)d_in[3];
  const float* att_dst = (const float*)d_in[4];
  const float* bias    = (const float*)d_in[5];
  const float* lin_w   = (const float*)d_in[6];
  const float* lin_b   = (const float*)d_in[7];
  float*       out     = (float*)d_out;

  const int N  = in_sizes[0] / D_FEAT;
  const int E  = in_sizes[1] / 2;
  const int Et = E + N;
  const int* srcv = ei;
  const int* dstv = ei + E;

  // Workspace layout: [agg | emax | denom] (zeroed together) | xp | a_src | a_dst | ebuf
  char* ws = (char*)d_ws;
  float*    agg   = (float*)ws;    ws += (size_t)N * C_TOT * sizeof(float);
  unsigned* emax  = (unsigned*)ws; ws += (size_t)N * H_HEADS * sizeof(unsigned);
  float*    denom = (float*)ws;    ws += (size_t)N * H_HEADS * sizeof(float);
  float*    xp    = (float*)ws;    ws += (size_t)N * C_TOT * sizeof(float);
  float*    a_src = (float*)ws;    ws += (size_t)N * H_HEADS * sizeof(float);
  float*    a_dst = (float*)ws;    ws += (size_t)N * H_HEADS * sizeof(float);
  float*    ebuf  = (float*)ws;

  const size_t zero_elems = (size_t)N * C_TOT + 2u * (size_t)N * H_HEADS;
  fill_zero4<<<2048, 256, 0, stream>>>((uint4*)agg, zero_elems / 4);

  dim3 gProj((N + 15) / 16, (C_TOT + 127) / 128);
  proj_wmma<<<gProj, 256, 0, stream>>>(x, Wf, xp, N);

  const int nWavesAtt = N * H_HEADS;
  att_logits<<<(nWavesAtt * 32 + 255) / 256, 256, 0, stream>>>(xp, att_src, att_dst,
                                                               a_src, a_dst, N);

  edge_logits<<<(Et + 255) / 256, 256, 0, stream>>>(srcv, dstv, a_src, a_dst,
                                                    ebuf, emax, E, Et);
  edge_exp<<<(Et + 255) / 256, 256, 0, stream>>>(dstv, ebuf, emax, denom, E, Et);
  edge_scatter<<<Et, 160, 0, stream>>>(srcv, dstv, xp, ebuf, denom, agg, E, Et);

  finalize<<<(N * 32 + 255) / 256, 256, 0, stream>>>(agg, bias, lin_w, lin_b, out, N);
}